// Frame_52218212385438
// MI455X (gfx1250) — compile-verified
//
#include <hip/hip_runtime.h>

// Frame: patch-extract (non-overlapping, stride==patch) + dense linear + ReLU
//   => GEMM  M=131072, N=256, K=256, fp32, out = relu(A * W^T + b)
// A row m = contiguous 256 floats of x starting at m*256, permuted (p,c)->(c*64+p).
//
// Design (MI455X, wave32, WMMA):
//  - memory-bound: ~268 MB of mandatory HBM traffic -> ~11.5 us floor @ 23.3 TB/s;
//    fp32 V_WMMA_F32_16X16X4_F32 keeps full reference precision and still beats
//    the memory floor, so no downconvert.
//  - wave w of 16 owns N-tile [16w,16w+16); its whole B slice (256x16) lives in
//    128 VGPRs (64 x float2 fragments), loaded once from L2-resident W.
//  - A tiles (16x256) staged to LDS with the patch transpose applied; 260-dword
//    row stride makes the per-lane b64 fragment loads hit all 64 banks once.
//  - TWO independent accumulator chains (even/odd k-step) break the WMMA D->C
//    serial dependency so the matrix pipe can stay 2-deep per wave on top of
//    4-wave/SIMD interleaving.

typedef __attribute__((ext_vector_type(2))) float v2f;
typedef __attribute__((ext_vector_type(4))) float v4f;
typedef __attribute__((ext_vector_type(8))) float v8f;

#define KD 256              // K (= D)
#define ND 256              // N (= D)
#define MT 16               // rows per M-tile (one WMMA M)
#define MTILES 8192         // 131072 / 16
#define LDS_STRIDE 260      // 256 + 4 pad dwords -> conflict-free b64 frag loads
#define THREADS 512         // 16 waves (wave32): one wave per 16-wide N-tile
#define NWG 1024

__global__ __launch_bounds__(THREADS, 1)
void patch_linear_relu_wmma(const float* __restrict__ x,
                            const float* __restrict__ W,
                            const float* __restrict__ bias,
                            float* __restrict__ out)
{
    __shared__ float ldsA[MT * LDS_STRIDE];   // ~16.6 KB of the 320 KB WGP LDS

    const int t     = threadIdx.x;
    const int wv    = t >> 5;            // wave id 0..15 -> owns N-tile [wv*16, wv*16+16)
    const int lane  = t & 31;
    const int nrow  = lane & 15;         // N (and A-row M) index within tile
    const int khalf = (lane >> 4) * 2;   // K sub-offset: lanes 16-31 hold K+2,K+3

    // ---------------------------------------------------------------
    // Register-resident B slice (K=256 x N=16) for this wave.
    // v_wmma_f32_16x16x4_f32 B fragment (mirrors the ISA 16x4 A layout):
    //   v0: lanes0-15 = B[k0+0][n], lanes16-31 = B[k0+2][n]
    //   v1: lanes0-15 = B[k0+1][n], lanes16-31 = B[k0+3][n]
    // with B[k][n] = W[wv*16+n][k]  (W row-major, row stride 256).
    // ---------------------------------------------------------------
    const float* wrow = W + (size_t)(wv * 16 + nrow) * KD + khalf;
    v2f bfrag[64];
#pragma unroll
    for (int kk = 0; kk < 64; ++kk)
        bfrag[kk] = *(const v2f*)(wrow + kk * 4);

    const float bval = bias[wv * 16 + nrow];

    for (int mt = blockIdx.x; mt < MTILES; mt += NWG) {
        // -----------------------------------------------------------
        // Stage A tile: 16 rows x 256 K = 4096 contiguous floats.
        // Coalesced b128 loads; scatter into LDS with (p,c)->(c*64+p)
        // transpose and +4-dword row padding.
        // -----------------------------------------------------------
        const float* src = x + (size_t)mt * (MT * KD);
#pragma unroll
        for (int i = 0; i < 2; ++i) {
            const int e4  = t + i * THREADS;      // float4 index 0..1023
            const v4f v   = *(const v4f*)(src + (size_t)e4 * 4);
            const int row = e4 >> 6;              // 64 float4 per A row
            const int p   = e4 & 63;              // position within patch
            float* d = &ldsA[row * LDS_STRIDE + p];
            d[0]   = v.x;                         // c = 0
            d[64]  = v.y;                         // c = 1
            d[128] = v.z;                         // c = 2
            d[192] = v.w;                         // c = 3
        }
        __syncthreads();

        // Prefetch next M-tile's A block toward L2 while we compute.
        if (mt + NWG < MTILES && t < 128) {
            const char* nsrc = (const char*)(x + (size_t)(mt + NWG) * (MT * KD));
            __builtin_prefetch(nsrc + t * 128, 0, 0);
        }

        // -----------------------------------------------------------
        // K=256 as 64 x v_wmma_f32_16x16x4_f32 split across two
        // independent accumulator chains (even/odd k-step) to break
        // the D->C serial dependency. Bias seeds chain 0.
        // -----------------------------------------------------------
        v8f acc0 = { bval, bval, bval, bval, bval, bval, bval, bval };
        v8f acc1 = { 0.f, 0.f, 0.f, 0.f, 0.f, 0.f, 0.f, 0.f };
        const float* arow = &ldsA[nrow * LDS_STRIDE + khalf];
#pragma unroll
        for (int kk = 0; kk < 32; ++kk) {
            const v2f a0 = *(const v2f*)(arow + (2 * kk)     * 4);
            const v2f a1 = *(const v2f*)(arow + (2 * kk + 1) * 4);
            acc0 = __builtin_amdgcn_wmma_f32_16x16x4_f32(
                       false, a0, false, bfrag[2 * kk],
                       (short)0, acc0, false, false);
            acc1 = __builtin_amdgcn_wmma_f32_16x16x4_f32(
                       false, a1, false, bfrag[2 * kk + 1],
                       (short)0, acc1, false, false);
        }

        // -----------------------------------------------------------
        // Merge chains + ReLU + store. D layout: lane l, vgpr g ->
        // M = g + (l>>4)*8, N = l&15. 64B row-contiguous stores per
        // half-wave.
        // -----------------------------------------------------------
        float* obase = out + (size_t)mt * MT * ND + wv * 16 + nrow;
        const int mb = (lane >> 4) * 8;
#pragma unroll
        for (int g = 0; g < 8; ++g) {
            float vv = acc0[g] + acc1[g];
            vv = vv > 0.0f ? vv : 0.0f;
            obase[(size_t)(g + mb) * ND] = vv;
        }
        __syncthreads();   // protect ldsA before next iteration's staging
    }
}

extern "C" void kernel_launch(void* const* d_in, const int* in_sizes, int n_in,
                              void* d_out, int out_size, void* d_ws, size_t ws_size,
                              hipStream_t stream) {
    const float* x = (const float*)d_in[0];   // (256, 32768, 4) fp32
    const float* W = (const float*)d_in[1];   // (256, 256) fp32
    const float* b = (const float*)d_in[2];   // (256,) fp32
    float* out = (float*)d_out;               // (256, 512, 256) fp32

    patch_linear_relu_wmma<<<dim3(NWG), dim3(THREADS), 0, stream>>>(x, W, b, out);
}